// GGNN_32624571580955
// MI455X (gfx1250) — compile-verified
//
#include <hip/hip_runtime.h>
#include <hip/hip_bf16.h>

typedef __attribute__((ext_vector_type(2))) float v2f;
typedef __attribute__((ext_vector_type(8))) float v8f;

#define F 128
#define LDA 132   // LDS row stride (padded to avoid 64-bank conflicts)

__device__ __forceinline__ float sigmoidf_(float x) {
    return 1.0f / (1.0f + __expf(-x));
}

// ---------------------------------------------------------------------------
// m = h @ W     (W row-major [128,128])
// One block = 8 waves; block handles 16 rows; wave w handles cols [16w,16w+16)
// fp32 WMMA 16x16x4, K-loop of 32 steps.
// ---------------------------------------------------------------------------
__global__ __launch_bounds__(256) void ggnn_gemm_m(
    const float* __restrict__ A, const float* __restrict__ W,
    float* __restrict__ C, int nrows)
{
    __shared__ float As[16 * LDA];
    const int tid  = threadIdx.x;
    const int row0 = blockIdx.x * 16;

    for (int i = tid; i < 16 * F; i += 256) {
        int r = i >> 7, c = i & (F - 1);
        int gr = row0 + r;
        As[r * LDA + c] = (gr < nrows) ? A[(size_t)gr * F + c] : 0.0f;
    }
    __syncthreads();

    const int wave  = tid >> 5;
    const int lane  = tid & 31;
    const int col0  = wave * 16;
    const int mrow  = lane & 15;         // A row / B col within tile
    const int khalf = (lane >> 4) * 2;   // 0 for lanes 0-15, 2 for lanes 16-31

    v8f acc = {};
    for (int k = 0; k < F; k += 4) {
        v2f a, b;
        a.x = As[mrow * LDA + k + khalf];
        a.y = As[mrow * LDA + k + khalf + 1];
        b.x = W[(k + khalf) * F + col0 + mrow];
        b.y = W[(k + khalf + 1) * F + col0 + mrow];
        acc = __builtin_amdgcn_wmma_f32_16x16x4_f32(
            false, a, false, b, (short)0, acc, false, false);
    }

    const int rbase = row0 + ((lane >> 4) << 3);
    const int c = col0 + (lane & 15);
#pragma unroll
    for (int v = 0; v < 8; ++v) {
        int gr = rbase + v;
        if (gr < nrows) C[(size_t)gr * F + c] = acc[v];
    }
}

// ---------------------------------------------------------------------------
// agg = 0
// ---------------------------------------------------------------------------
__global__ __launch_bounds__(256) void ggnn_zero(float* __restrict__ p, int n)
{
    int i = blockIdx.x * 256 + threadIdx.x;
    int stride = gridDim.x * 256;
    for (; i < n; i += stride) p[i] = 0.0f;
}

// ---------------------------------------------------------------------------
// agg[dst] += m[src]   (one wave per edge, 4 floats per lane)
// ---------------------------------------------------------------------------
__global__ __launch_bounds__(256) void ggnn_scatter(
    const float* __restrict__ m, const int* __restrict__ src,
    const int* __restrict__ dst, float* __restrict__ agg, int nedges)
{
    int t = blockIdx.x * 256 + threadIdx.x;
    int e = t >> 5;
    if (e >= nedges) return;
    int lane = t & 31;
    int s = src[e], d = dst[e];
    const float4 v = *(const float4*)(m + (size_t)s * F + lane * 4);
    float* ap = agg + (size_t)d * F + lane * 4;
    atomicAdd(ap + 0, v.x);
    atomicAdd(ap + 1, v.y);
    atomicAdd(ap + 2, v.z);
    atomicAdd(ap + 3, v.w);
}

// ---------------------------------------------------------------------------
// GRU cell, in-place h update.
// gi = agg @ w_ih^T + b_ih ; gh = h @ w_hh^T + b_hh   (w_* are [384,128])
// Block handles 16 rows; wave w handles feature cols [16w,16w+16).
// 6 accumulators (r/z/n for gi and gh) -> 6 WMMAs per K step.
// ---------------------------------------------------------------------------
__global__ __launch_bounds__(256) void ggnn_gru(
    const float* __restrict__ agg, float* __restrict__ h,
    const float* __restrict__ w_ih, const float* __restrict__ w_hh,
    const float* __restrict__ b_ih, const float* __restrict__ b_hh,
    int nrows)
{
    __shared__ float Ag[16 * LDA];
    __shared__ float Hs[16 * LDA];
    const int tid  = threadIdx.x;
    const int row0 = blockIdx.x * 16;

    for (int i = tid; i < 16 * F; i += 256) {
        int r = i >> 7, c = i & (F - 1);
        int gr = row0 + r;
        bool ok = (gr < nrows);
        Ag[r * LDA + c] = ok ? agg[(size_t)gr * F + c] : 0.0f;
        Hs[r * LDA + c] = ok ? h[(size_t)gr * F + c] : 0.0f;
    }
    __syncthreads();

    const int wave  = tid >> 5;
    const int lane  = tid & 31;
    const int col0  = wave * 16;
    const int mrow  = lane & 15;
    const int khalf = (lane >> 4) * 2;
    const int jr = 0 * F + col0 + mrow;   // r-gate output col in [0,384)
    const int jz = 1 * F + col0 + mrow;
    const int jn = 2 * F + col0 + mrow;

    v8f air = {}, aiz = {}, ain = {};
    v8f ahr = {}, ahz = {}, ahn = {};

    for (int k = 0; k < F; k += 4) {
        v2f ag, ah;
        ag.x = Ag[mrow * LDA + k + khalf];
        ag.y = Ag[mrow * LDA + k + khalf + 1];
        ah.x = Hs[mrow * LDA + k + khalf];
        ah.y = Hs[mrow * LDA + k + khalf + 1];
        // B[k][j] = w[j*128 + k]  -> consecutive k, load as float2
        v2f br_i = *(const v2f*)(w_ih + (size_t)jr * F + k + khalf);
        v2f bz_i = *(const v2f*)(w_ih + (size_t)jz * F + k + khalf);
        v2f bn_i = *(const v2f*)(w_ih + (size_t)jn * F + k + khalf);
        v2f br_h = *(const v2f*)(w_hh + (size_t)jr * F + k + khalf);
        v2f bz_h = *(const v2f*)(w_hh + (size_t)jz * F + k + khalf);
        v2f bn_h = *(const v2f*)(w_hh + (size_t)jn * F + k + khalf);
        air = __builtin_amdgcn_wmma_f32_16x16x4_f32(false, ag, false, br_i, (short)0, air, false, false);
        aiz = __builtin_amdgcn_wmma_f32_16x16x4_f32(false, ag, false, bz_i, (short)0, aiz, false, false);
        ain = __builtin_amdgcn_wmma_f32_16x16x4_f32(false, ag, false, bn_i, (short)0, ain, false, false);
        ahr = __builtin_amdgcn_wmma_f32_16x16x4_f32(false, ah, false, br_h, (short)0, ahr, false, false);
        ahz = __builtin_amdgcn_wmma_f32_16x16x4_f32(false, ah, false, bz_h, (short)0, ahz, false, false);
        ahn = __builtin_amdgcn_wmma_f32_16x16x4_f32(false, ah, false, bn_h, (short)0, ahn, false, false);
    }

    // Gate math in the C/D layout: VGPR v holds row = v + 8*(lane>=16), col = lane&15
    const int rloc  = ((lane >> 4) << 3);
    const int c     = col0 + (lane & 15);
    const float bir = b_ih[0 * F + c], bhr = b_hh[0 * F + c];
    const float biz = b_ih[1 * F + c], bhz = b_hh[1 * F + c];
    const float bin = b_ih[2 * F + c], bhn = b_hh[2 * F + c];

#pragma unroll
    for (int v = 0; v < 8; ++v) {
        int lr = rloc + v;
        int gr = row0 + lr;
        float r = sigmoidf_((air[v] + bir) + (ahr[v] + bhr));
        float z = sigmoidf_((aiz[v] + biz) + (ahz[v] + bhz));
        float n = tanhf((ain[v] + bin) + r * (ahn[v] + bhn));
        float hv = Hs[lr * LDA + c];
        if (gr < nrows) h[(size_t)gr * F + c] = (1.0f - z) * n + z * hv;
    }
}

// ---------------------------------------------------------------------------
// out[n] = relu(h[n]) . lin_w + lin_b     (one wave per node)
// ---------------------------------------------------------------------------
__global__ __launch_bounds__(256) void ggnn_proj(
    const float* __restrict__ h, const float* __restrict__ lin_w,
    const float* __restrict__ lin_b, float* __restrict__ out, int nrows)
{
    int wave = threadIdx.x >> 5;
    int lane = threadIdx.x & 31;
    int node = blockIdx.x * 8 + wave;
    if (node >= nrows) return;
    const float4 hv = *(const float4*)(h + (size_t)node * F + lane * 4);
    const float4 wv = *(const float4*)(lin_w + lane * 4);
    float s = fmaxf(hv.x, 0.0f) * wv.x + fmaxf(hv.y, 0.0f) * wv.y +
              fmaxf(hv.z, 0.0f) * wv.z + fmaxf(hv.w, 0.0f) * wv.w;
#pragma unroll
    for (int off = 16; off > 0; off >>= 1) s += __shfl_xor(s, off, 32);
    if (lane == 0) out[node] = s + lin_b[0];
}

// ---------------------------------------------------------------------------
extern "C" void kernel_launch(void* const* d_in, const int* in_sizes, int n_in,
                              void* d_out, int out_size, void* d_ws, size_t ws_size,
                              hipStream_t stream)
{
    const float* x      = (const float*)d_in[0];
    const int*   eidx   = (const int*)  d_in[1];   // [2, E] flattened
    const float* weight = (const float*)d_in[2];   // [5,128,128]
    const float* w_ih   = (const float*)d_in[3];   // [384,128]
    const float* w_hh   = (const float*)d_in[4];   // [384,128]
    const float* b_ih   = (const float*)d_in[5];   // [384]
    const float* b_hh   = (const float*)d_in[6];   // [384]
    const float* lin_w  = (const float*)d_in[7];   // [1,128]
    const float* lin_b  = (const float*)d_in[8];   // [1]
    float* out = (float*)d_out;

    const int N = in_sizes[0] / F;
    const int E = in_sizes[1] / 2;

    float* h   = (float*)d_ws;
    float* m   = h + (size_t)N * F;
    float* agg = m + (size_t)N * F;

    hipMemcpyAsync(h, x, (size_t)N * F * sizeof(float),
                   hipMemcpyDeviceToDevice, stream);

    const int rowTiles     = (N + 15) / 16;
    const int zeroBlocks   = ((N * F) + 255) / 256;        // grid-stride anyway
    const int scatBlocks   = (E * 32 + 255) / 256;

    for (int l = 0; l < 5; ++l) {
        ggnn_gemm_m<<<rowTiles, 256, 0, stream>>>(
            h, weight + (size_t)l * F * F, m, N);
        ggnn_zero<<<zeroBlocks < 4096 ? zeroBlocks : 4096, 256, 0, stream>>>(
            agg, N * F);
        ggnn_scatter<<<scatBlocks, 256, 0, stream>>>(
            m, eidx, eidx + E, agg, E);
        ggnn_gru<<<rowTiles, 256, 0, stream>>>(
            agg, h, w_ih, w_hh, b_ih, b_hh, N);
    }

    ggnn_proj<<<(N + 7) / 8, 256, 0, stream>>>(h, lin_w, lin_b, out, N);
}